// PointNet_71347996721271
// MI455X (gfx1250) — compile-verified
//
#include <hip/hip_runtime.h>

#define NG   1024   // graphs
#define NP   128    // points per graph
#define KNN  16     // neighbors
#define CH   32     // hidden channels
#define CHP  33     // LDS row stride (bank-conflict padding)
#define COUT 40

typedef float v2f __attribute__((ext_vector_type(2)));
typedef float v8f __attribute__((ext_vector_type(8)));

__device__ __forceinline__ v8f wmma4(v2f a, v2f b, v8f c) {
    // D(16x16,f32) = A(16x4,f32) x B(4x16,f32) + C  -- native fp32 WMMA on CDNA5
    return __builtin_amdgcn_wmma_f32_16x16x4_f32(
        false, a, false, b, (short)0, c, false, false);
}

// Preload B-tiles (two 16-col tiles per K-chunk) from a stride-CHP LDS matrix
// into registers. Loop-invariant across the per-point loops; hoisted manually
// because the in-loop __syncthreads blocks compiler hoisting of LDS loads.
template <int NCH>
__device__ __forceinline__ void load_wtiles(const float* W, int m, int half,
                                            v2f* w0, v2f* w1) {
#pragma unroll
    for (int kk = 0; kk < NCH; ++kk) {
        const int ks = kk * 4 + half * 2;
        w0[kk].x = W[ks * CHP + m];        w0[kk].y = W[(ks + 1) * CHP + m];
        w1[kk].x = W[ks * CHP + 16 + m];   w1[kk].y = W[(ks + 1) * CHP + 16 + m];
    }
}

// GEMM2: out(16x32) = hid(16x32) @ W(32x32) + b, max over 16 edge-rows, ReLU.
// hid in LDS (stride CHP); W pre-staged in registers. Lane L owns channel L.
__device__ __forceinline__ float gemm2_maxpool(const float* hl, const v2f* w0,
                                               const v2f* w1, float bb0, float bb1,
                                               int m, int half) {
    v8f d0, d1;
#pragma unroll
    for (int r = 0; r < 8; ++r) { d0[r] = bb0; d1[r] = bb1; }
#pragma unroll
    for (int kk = 0; kk < 8; ++kk) {
        const int ks = kk * 4 + half * 2;
        v2f aa; aa.x = hl[m * CHP + ks]; aa.y = hl[m * CHP + ks + 1];
        d0 = wmma4(aa, w0[kk], d0);
        d1 = wmma4(aa, w1[kk], d1);
    }
    float p0 = d0[0], p1 = d1[0];
#pragma unroll
    for (int r = 1; r < 8; ++r) { p0 = fmaxf(p0, d0[r]); p1 = fmaxf(p1, d1[r]); }
    p0 = fmaxf(p0, __shfl_xor(p0, 16));          // combine row halves across lane halves
    p1 = fmaxf(p1, __shfl_xor(p1, 16));
    return fmaxf(half ? p1 : p0, 0.0f);          // outer ReLU
}

// ---------------------------------------------------------------------------
// Fully fused: 1 block = 1 graph (256 thr = 8 waves). kNN -> layer1 -> layer2
// -> global max pool (LDS atomics) -> classifier. All intermediates in LDS.
// ---------------------------------------------------------------------------
__global__ __launch_bounds__(256) void pointnet_fused(
        const float* __restrict__ pos,
        const float* __restrict__ W1a, const float* __restrict__ b1a,
        const float* __restrict__ W1b, const float* __restrict__ b1b,
        const float* __restrict__ W2a, const float* __restrict__ b2a,
        const float* __restrict__ W2b, const float* __restrict__ b2b,
        const float* __restrict__ Wc,  const float* __restrict__ bc,
        float* __restrict__ out) {
    __shared__ float sPx[NP], sPy[NP];
    __shared__ int   sIdx[NP * KNN];
    __shared__ float sH1[NP * CHP];
    __shared__ float sW1a[4 * CHP],  sB1a[CH];
    __shared__ float sW1b[CH * CHP], sB1b[CH];
    __shared__ float sW2a[34 * CHP], sB2a[CH];
    __shared__ float sW2b[CH * CHP], sB2b[CH];
    __shared__ float sWc[CH * COUT], sBc[COUT];
    __shared__ float sHid[8][16 * CHP];          // per-wave 16x32 scratch tile
    __shared__ int   sPool[CH];

    const int t = threadIdx.x, g = blockIdx.x;
    const int lane = t & 31, wave = t >> 5;
    const int m = lane & 15, half = lane >> 4;

    // -------- stage 0: stage positions + all weights into LDS --------------
    if (t < NP) { sPx[t] = pos[(g * NP + t) * 2]; sPy[t] = pos[(g * NP + t) * 2 + 1]; }
    for (int i = t; i < 4 * CH;    i += 256) sW1a[(i >> 5) * CHP + (i & 31)] = W1a[i];
    for (int i = t; i < CH * CH;   i += 256) sW1b[(i >> 5) * CHP + (i & 31)] = W1b[i];
    for (int i = t; i < 34 * CH;   i += 256) sW2a[(i >> 5) * CHP + (i & 31)] = W2a[i];
    for (int i = t; i < CH * CH;   i += 256) sW2b[(i >> 5) * CHP + (i & 31)] = W2b[i];
    for (int i = t; i < CH * COUT; i += 256) sWc[i] = Wc[i];
    if (t < CH)   { sB1a[t] = b1a[t]; sB1b[t] = b1b[t];
                    sB2a[t] = b2a[t]; sB2b[t] = b2b[t]; sPool[t] = 0; }
    if (t < COUT) sBc[t] = bc[t];
    __syncthreads();

    // -------- stage 1: kNN, k=16 incl. self (threads 0..127, 1 row each) ---
    if (t < NP) {
        const float xi = sPx[t], yi = sPy[t];
        float bd[KNN]; int bi[KNN];
#pragma unroll
        for (int k = 0; k < KNN; ++k) { bd[k] = 3.4e38f; bi[k] = 0; }
        float worst = 3.4e38f; int wslot = 0;
        for (int j = 0; j < NP; ++j) {
            const float dx = sPx[j] - xi, dy = sPy[j] - yi;
            const float d2 = dx * dx + dy * dy;
            if (d2 < worst) {
#pragma unroll
                for (int k = 0; k < KNN; ++k)
                    if (k == wslot) { bd[k] = d2; bi[k] = j; }
                worst = bd[0]; wslot = 0;
#pragma unroll
                for (int k = 1; k < KNN; ++k)
                    if (bd[k] > worst) { worst = bd[k]; wslot = k; }
            }
        }
#pragma unroll
        for (int k = 0; k < KNN; ++k) sIdx[t * KNN + k] = bi[k];
    }
    __syncthreads();

    float* hl = sHid[wave];

    // -------- stage 2: layer 1 (each wave: 16 points, 1 M-tile per point) --
    // msg = [pos_j(2), pos_j - pos_i(2)] : K=4 -> single WMMA K-chunk
    {
        const int r0 = half * 2;                 // hoisted weights + biases
        v2f wA0, wA1;
        wA0.x = sW1a[r0 * CHP + m];      wA0.y = sW1a[(r0 + 1) * CHP + m];
        wA1.x = sW1a[r0 * CHP + 16 + m]; wA1.y = sW1a[(r0 + 1) * CHP + 16 + m];
        v2f wB0[8], wB1[8];
        load_wtiles<8>(sW1b, m, half, wB0, wB1);
        const float ba0 = sB1a[m], ba1 = sB1a[16 + m];
        const float bb0 = sB1b[m], bb1 = sB1b[16 + m];

        for (int it = 0; it < 16; ++it) {
            const int p = wave * 16 + it;
            const int j = sIdx[p * KNN + m];
            const float pjx = sPx[j], pjy = sPy[j];
            v2f a;
            a.x = half ? (pjx - sPx[p]) : pjx;
            a.y = half ? (pjy - sPy[p]) : pjy;
            v8f c0, c1;
#pragma unroll
            for (int r = 0; r < 8; ++r) { c0[r] = ba0; c1[r] = ba1; }
            c0 = wmma4(a, wA0, c0);
            c1 = wmma4(a, wA1, c1);
#pragma unroll
            for (int r = 0; r < 8; ++r) {        // ReLU + C/D-layout -> row-major
                const int row = r + 8 * half;
                hl[row * CHP + m]      = fmaxf(c0[r], 0.0f);
                hl[row * CHP + 16 + m] = fmaxf(c1[r], 0.0f);
            }
            __syncthreads();                     // uniform: all waves, 16 iters
            const float v = gemm2_maxpool(hl, wB0, wB1, bb0, bb1, m, half);
            sH1[p * CHP + lane] = v;             // lane L -> channel L
        }
    }
    __syncthreads();

    // -------- stage 3: layer 2 + fused max-pool (LDS int atomics) ----------
    // msg = [h_j(32), rel(2)] : K=34 zero-padded to 36 -> 9 WMMA K-chunks
    {
        v2f wA0[9], wA1[9];                      // hoisted weights + biases
        load_wtiles<8>(sW2a, m, half, wA0, wA1);
        wA0[8].x = half ? 0.0f : sW2a[32 * CHP + m];        // chunk 8: rows {32,33},
        wA0[8].y = half ? 0.0f : sW2a[33 * CHP + m];        // rows {34,35} = 0 pad
        wA1[8].x = half ? 0.0f : sW2a[32 * CHP + 16 + m];
        wA1[8].y = half ? 0.0f : sW2a[33 * CHP + 16 + m];
        v2f wB0[8], wB1[8];
        load_wtiles<8>(sW2b, m, half, wB0, wB1);
        const float ba0 = sB2a[m], ba1 = sB2a[16 + m];
        const float bb0 = sB2b[m], bb1 = sB2b[16 + m];

        for (int it = 0; it < 16; ++it) {
            const int p = wave * 16 + it;
            const int j = sIdx[p * KNN + m];
            const float rx = sPx[j] - sPx[p], ry = sPy[j] - sPy[p];
            v8f c0, c1;
#pragma unroll
            for (int r = 0; r < 8; ++r) { c0[r] = ba0; c1[r] = ba1; }
#pragma unroll
            for (int kk = 0; kk < 8; ++kk) {     // h_j channels 0..31
                const int ks = kk * 4 + half * 2;
                v2f aa; aa.x = sH1[j * CHP + ks]; aa.y = sH1[j * CHP + ks + 1];
                c0 = wmma4(aa, wA0[kk], c0);
                c1 = wmma4(aa, wA1[kk], c1);
            }
            {                                    // chunk 8: feats {32,33}=rel
                v2f aa; aa.x = half ? 0.0f : rx; aa.y = half ? 0.0f : ry;
                c0 = wmma4(aa, wA0[8], c0);
                c1 = wmma4(aa, wA1[8], c1);
            }
#pragma unroll
            for (int r = 0; r < 8; ++r) {
                const int row = r + 8 * half;
                hl[row * CHP + m]      = fmaxf(c0[r], 0.0f);
                hl[row * CHP + 16 + m] = fmaxf(c1[r], 0.0f);
            }
            __syncthreads();
            const float v = gemm2_maxpool(hl, wB0, wB1, bb0, bb1, m, half);
            // ReLU output >= 0: int ordering == float ordering; pool pre-zeroed
            atomicMax(&sPool[lane], __float_as_int(v));
        }
    }
    __syncthreads();

    // -------- stage 4: classifier out[g] = pool @ Wc + bc ------------------
    if (t < COUT) {
        float acc = sBc[t];
#pragma unroll
        for (int c = 0; c < CH; ++c)
            acc += __int_as_float(sPool[c]) * sWc[c * COUT + t];
        out[g * COUT + t] = acc;
    }
}

extern "C" void kernel_launch(void* const* d_in, const int* in_sizes, int n_in,
                              void* d_out, int out_size, void* d_ws, size_t ws_size,
                              hipStream_t stream) {
    (void)in_sizes; (void)n_in; (void)out_size; (void)d_ws; (void)ws_size;
    const float* pos = (const float*)d_in[0];
    /* d_in[1] = batch (unused: uniform G*P layout) */
    pointnet_fused<<<NG, 256, 0, stream>>>(
        pos,
        (const float*)d_in[2],  (const float*)d_in[3],
        (const float*)d_in[4],  (const float*)d_in[5],
        (const float*)d_in[6],  (const float*)d_in[7],
        (const float*)d_in[8],  (const float*)d_in[9],
        (const float*)d_in[10], (const float*)d_in[11],
        (float*)d_out);
}